// STU_24867860644438
// MI455X (gfx1250) — compile-verified
//
#include <hip/hip_runtime.h>

// MI455X (gfx1250) STU kernel.
// Math: out[b,t,d] = sum_{s even <= t} 2*phi_proj[s,d]*x_proj[b,t-s,d]
// (the rfft(8192) pipeline is an exact linear conv; the sgn trick cancels odd lags).
// K0: pack Mi -> P[k/4][n][4] so WMMA B-frags are single b64 loads        (2.3 MB pass)
// K1: effective half-rate filter he[j,d] = 2*(phi@M_filters)[2j,d]        (tiny, VALU)
// K2: x_proj = x @ M_inputs via V_WMMA_F32_16X16X4_F32, 32x64 per wave    (19.3 GFLOP)
// K3: two half-rate depthwise causal convs, LDS sliding window (b128
//     staging, unroll-8 register-renamed window) + global_prefetch        (26.6 GFLOP, L2-resident)

typedef __attribute__((ext_vector_type(2))) float v2f;
typedef __attribute__((ext_vector_type(8))) float v8f;

#define B_ 4
#define S_ 4096
#define D_ 768
#define K_ 24
#define HALF_S 2048
#define K4_ (D_ / 4)   // 192 k-groups

// ---------------- K0: pack Mi for b64 B-fragment loads ----------------
// P[((k4*D_)+n)*4 + c] = Mi[(4*k4+c)*D_ + n]
__global__ __launch_bounds__(256)
void stu_pack_kernel(const float* __restrict__ Mi, float* __restrict__ P) {
    int n  = blockIdx.x * 256 + threadIdx.x;   // grid.x = 3
    int k4 = blockIdx.y;                       // 0..191
    if (n >= D_) return;
    float4 v;
    v.x = Mi[(size_t)(4 * k4 + 0) * D_ + n];
    v.y = Mi[(size_t)(4 * k4 + 1) * D_ + n];
    v.z = Mi[(size_t)(4 * k4 + 2) * D_ + n];
    v.w = Mi[(size_t)(4 * k4 + 3) * D_ + n];
    *(float4*)(P + ((size_t)k4 * D_ + n) * 4) = v;   // 16B aligned
}

// ---------------- K1: effective filter ----------------
__global__ __launch_bounds__(256)
void stu_filter_kernel(const float* __restrict__ phi,   // S x K
                       const float* __restrict__ Mf,    // K x D
                       float* __restrict__ he) {        // HALF_S x D
    int d = blockIdx.x * 256 + threadIdx.x;             // grid.x = 3
    int j = blockIdx.y;                                 // 0..2047
    if (d >= D_) return;
    const float* prow = phi + (2 * j) * K_;             // even rows only
    float acc = 0.f;
#pragma unroll
    for (int k = 0; k < K_; ++k)
        acc = fmaf(prow[k], Mf[k * D_ + d], acc);
    he[j * D_ + d] = 2.0f * acc;
}

// ---------------- K2: x_proj GEMM via f32 WMMA ----------------
// C(16384x768) = X(16384x768) @ Mi(768x768). Wave computes a 32x64 strip.
// A frag (16x4 f32): lane<16 holds A[lane, k..k+1]; lane>=16 holds A[lane-16, k+2..k+3].
// B frag (4x16 f32): lane group g=lane>>4 holds rows k+2g, k+2g+1 at column (lane&15)
//                    -> one v2f load from packed P.
// C (16x16 f32):     vgpr v -> row v (lanes 0-15) / row v+8 (lanes 16-31).
__global__ __launch_bounds__(128)
void stu_xproj_wmma(const float* __restrict__ X,
                    const float* __restrict__ P,    // packed Mi: [K4_][D_][4]
                    float* __restrict__ XP) {
    const int wave = threadIdx.x >> 5;
    const int lane = threadIdx.x & 31;
    const int half = lane >> 4;        // 0 or 1
    const int l16  = lane & 15;
    const int m0 = blockIdx.x * 32;              // 512 row tiles
    const int n0 = blockIdx.y * 256 + wave * 64; // 3 * 4 waves * 64 cols

    v8f c0[4] = {}, c1[4] = {};
    const float* arow0 = X + (size_t)(m0 + l16) * D_ + half * 2;
    const float* arow1 = arow0 + (size_t)16 * D_;
    const float* pcol  = P + ((size_t)(n0 + l16)) * 4 + 2 * half;

    for (int k4 = 0; k4 < K4_; ++k4) {
        v2f a0 = *(const v2f*)(arow0 + k4 * 4);   // 8B aligned
        v2f a1 = *(const v2f*)(arow1 + k4 * 4);
        const float* pk = pcol + (size_t)k4 * D_ * 4;
#pragma unroll
        for (int nt = 0; nt < 4; ++nt) {
            v2f b = *(const v2f*)(pk + nt * 64);  // 16 cols * 4 floats
            c0[nt] = __builtin_amdgcn_wmma_f32_16x16x4_f32(false, a0, false, b, (short)0, c0[nt], false, false);
            c1[nt] = __builtin_amdgcn_wmma_f32_16x16x4_f32(false, a1, false, b, (short)0, c1[nt], false, false);
        }
    }

    float* out0 = XP + (size_t)(m0 + half * 8) * D_ + n0 + l16;
    float* out1 = out0 + (size_t)16 * D_;
#pragma unroll
    for (int nt = 0; nt < 4; ++nt) {
#pragma unroll
        for (int v = 0; v < 8; ++v) {
            out0[(size_t)v * D_ + nt * 16] = c0[nt][v];
            out1[(size_t)v * D_ + nt * 16] = c1[nt][v];
        }
    }
}

// ---------------- K3: depthwise causal conv (per parity, half rate) ----------------
// y[i,d] = sum_{j<=i} he[j,d] * xc[i-j,d],  xc[i,d] = x_proj[b, 2i+p, d], i,j in [0,2048)
__global__ __launch_bounds__(256)
void stu_conv_kernel(const float* __restrict__ XP,  // (B*S) x D
                     const float* __restrict__ HE,  // HALF_S x D
                     float* __restrict__ OUT) {     // (B*S) x D
    __shared__ float sh_h[64 * 32];
    __shared__ float sh_x[128 * 32];

    const int ib = blockIdx.x;          // i-block: 32 blocks of 64 outputs
    const int d0 = blockIdx.y * 32;     // 24 channel tiles
    const int b  = blockIdx.z >> 1;     // batch
    const int p  = blockIdx.z & 1;      // parity
    const int tid  = threadIdx.x;
    const int lane = tid & 31;
    const int w    = tid >> 5;          // 8 waves -> 8 output groups
    const int i0 = ib * 64;
    const int d  = d0 + lane;

    float acc[8];
#pragma unroll
    for (int r = 0; r < 8; ++r) acc[r] = 0.f;

    const float* xpb = XP + (size_t)b * S_ * D_;

    for (int c = 0; c <= ib; ++c) {
        const int j0 = c * 64;
        // stage filter chunk he[j0..j0+63][d0..d0+31] as b128
        for (int idx = tid; idx < 64 * 8; idx += 256) {
            int jj = idx >> 3, q = idx & 7;
            *(float4*)(sh_h + jj * 32 + q * 4) =
                *(const float4*)(HE + (size_t)(j0 + jj) * D_ + d0 + q * 4);
        }
        // stage x window as b128: u = i0-j0-63 + s, s in [0,128); zero-pad OOB rows
        const int u_base = i0 - j0 - 63;
        for (int idx = tid; idx < 128 * 8; idx += 256) {
            int s = idx >> 3, q = idx & 7;
            int u = u_base + s;
            float4 v = {0.f, 0.f, 0.f, 0.f};
            if (u >= 0 && u < HALF_S)
                v = *(const float4*)(xpb + (size_t)(2 * u + p) * D_ + d0 + q * 4);
            *(float4*)(sh_x + s * 32 + q * 4) = v;
        }
        __syncthreads();

        // prefetch next chunk's lines into cache while we compute (counter-free)
        if (c < ib) {
            const int j0n = j0 + 64;
            __builtin_prefetch(&HE[(size_t)(j0n + (tid >> 2)) * D_ + d0 + (tid & 3) * 8], 0, 0);
            int u_n = i0 - j0n - 63 + (tid >> 1);
            if (u_n >= 0 && u_n < HALF_S)
                __builtin_prefetch(&xpb[(size_t)(2 * u_n + p) * D_ + d0 + (tid & 1) * 16], 0, 0);
        }

        // sliding window: xwin[r] tracks sh_x[8w + r + (63-jj)].
        // unroll-8: the 8 rotate-by-1 steps compose to a full window replacement,
        // so the compiler renames registers and emits zero v_mov in steady state.
        float xwin[8];
#pragma unroll
        for (int r = 0; r < 8; ++r) xwin[r] = sh_x[(8 * w + r) * 32 + lane];
#pragma unroll 8
        for (int jj = 63; jj >= 0; --jj) {
            float h = sh_h[jj * 32 + lane];
#pragma unroll
            for (int r = 0; r < 8; ++r) acc[r] = fmaf(h, xwin[r], acc[r]);
#pragma unroll
            for (int r = 0; r < 7; ++r) xwin[r] = xwin[r + 1];
            xwin[7] = sh_x[(8 * w + 71 - jj) * 32 + lane];  // max index 127: in range
        }
        __syncthreads();
    }

    // write out[b, 2*(i0+8w+r)+p, d]
#pragma unroll
    for (int r = 0; r < 8; ++r) {
        int t = 2 * (i0 + 8 * w + r) + p;
        OUT[((size_t)b * S_ + t) * D_ + d] = acc[r];
    }
}

extern "C" void kernel_launch(void* const* d_in, const int* in_sizes, int n_in,
                              void* d_out, int out_size, void* d_ws, size_t ws_size,
                              hipStream_t stream) {
    (void)in_sizes; (void)n_in; (void)out_size; (void)ws_size;
    const float* x   = (const float*)d_in[0];  // B,S,D
    const float* phi = (const float*)d_in[1];  // S,K
    const float* Mi  = (const float*)d_in[2];  // D,D
    const float* Mf  = (const float*)d_in[3];  // K,D
    float* out  = (float*)d_out;

    float* ws_x = (float*)d_ws;                            // 16384*768 f32 (50.3 MB)
    float* ws_h = ws_x + (size_t)B_ * S_ * D_;             // 2048*768  f32 ( 6.3 MB)
    float* ws_p = ws_h + (size_t)HALF_S * D_;              // 192*768*4 f32 ( 2.4 MB)

    stu_pack_kernel  <<<dim3(3, K4_, 1),            256, 0, stream>>>(Mi, ws_p);
    stu_filter_kernel<<<dim3(3, HALF_S, 1),         256, 0, stream>>>(phi, Mf, ws_h);
    stu_xproj_wmma   <<<dim3((B_ * S_) / 32, D_ / 256, 1), 128, 0, stream>>>(x, ws_p, ws_x);
    stu_conv_kernel  <<<dim3(HALF_S / 64, D_ / 32, B_ * 2), 256, 0, stream>>>(ws_x, ws_h, out);
}